// Abcnn1Portion_42417097017116
// MI455X (gfx1250) — compile-verified
//
#include <hip/hip_runtime.h>
#include <hip/hip_bf16.h>

typedef __attribute__((ext_vector_type(2))) float v2f;
typedef __attribute__((ext_vector_type(8))) float v8f;

#define B_ 16
#define S_ 256
#define D_ 256
#define SD 65536            // S_*D_
#define EPS_ATTN 1e-6f
#define EPS_BN   1e-5f

// workspace layout (float offsets)
#define OFF_N1     0                      // 4096 (B*S) row norms of x1
#define OFF_N2     4096                   // 4096 row norms of x2
#define OFF_ATT_T  8192                   // 1,048,576: attnT[b][i][j] = attn[b,j,i]
#define OFF_ATT_N  (8192 + 1048576)       // 1,048,576: attnN[b][j][i]
#define OFF_STATS  (8192 + 2*1048576)     // 8 floats: (sum,sumsq) x 4 channels

// ---------------- zero the stats accumulators ----------------
__global__ void k_zero(float* __restrict__ p, int n) {
    int i = blockIdx.x * blockDim.x + threadIdx.x;
    if (i < n) p[i] = 0.0f;
}

// ---------------- per-row squared norms of x1,x2 ----------------
// 8 waves/block, one wave per row of 256 floats
__global__ void k_norms(const float* __restrict__ x1, const float* __restrict__ x2,
                        float* __restrict__ norms) {
    const int wave = threadIdx.x >> 5;
    const int lane = threadIdx.x & 31;
    const int row  = blockIdx.x * 8 + wave;            // 0..8191
    const float* src = (row < B_*S_) ? (x1 + row * D_) : (x2 + (row - B_*S_) * D_);
    float s = 0.0f;
#pragma unroll
    for (int k = 0; k < 8; ++k) { float v = src[lane + 32 * k]; s += v * v; }
#pragma unroll
    for (int off = 16; off; off >>= 1) s += __shfl_xor(s, off, 32);
    if (lane == 0) norms[row] = s;
}

// ---------------- attn tiles via f32 WMMA ----------------
// grid (j_tile=16, i_tile=16, b=16), 1 wave/block -> 16x16 tile of G = x1 @ x2^T
__global__ void k_attn(const float* __restrict__ x1, const float* __restrict__ x2,
                       const float* __restrict__ norms,
                       float* __restrict__ attT, float* __restrict__ attN) {
    __shared__ float tile[16 * 17];
    const int b  = blockIdx.z;
    const int i0 = blockIdx.y * 16;
    const int j0 = blockIdx.x * 16;
    const int L    = threadIdx.x;
    const int lo   = L & 15;
    const int hi   = L >> 4;
    const int koff = hi * 2;

    const v2f* arow = (const v2f*)(x1 + (size_t)(b * S_ + i0 + lo) * D_); // A: lane=M(i)
    const v2f* brow = (const v2f*)(x2 + (size_t)(b * S_ + j0 + lo) * D_); // B: lane=N(j)

    v8f acc = {};
#pragma unroll 4
    for (int t = 0; t < 64; ++t) {
        v2f a = arow[(4 * t + koff) >> 1];
        v2f w = brow[(4 * t + koff) >> 1];
        acc = __builtin_amdgcn_wmma_f32_16x16x4_f32(false, a, false, w,
                                                    (short)0, acc, false, false);
    }
    // acc[r]: element (i = i0 + r + 8*hi, j = j0 + lo)
    const float* n1 = norms + OFF_N1;
    const float* n2 = norms + OFF_N2;
    const float nj = n2[b * S_ + j0 + lo];
#pragma unroll
    for (int r = 0; r < 8; ++r) {
        const int il = r + 8 * hi;
        const float ni = n1[b * S_ + i0 + il];
        float d2 = fmaxf(ni + nj - 2.0f * acc[r], 0.0f);
        tile[il * 17 + lo] = 1.0f / (1.0f + sqrtf(d2 + EPS_ATTN));
    }
    __syncthreads();
    // coalesced stores of both layouts
#pragma unroll
    for (int rr = 0; rr < 8; ++rr) {
        const int il = rr * 2 + hi;   // i row for attT
        attT[((size_t)(b * S_ + i0 + il)) * S_ + j0 + lo] = tile[il * 17 + lo];
        const int jl = rr * 2 + hi;   // j row for attN
        attN[((size_t)(b * S_ + j0 + jl)) * S_ + i0 + lo] = tile[lo * 17 + jl];
    }
}

// ---------------- attention GEMMs + bias + BN-stat accumulation ----------------
// C[b,m,o] = sum_k A[b,m,k] * W[o,k] + bias[o]
// z = b*2 + which;  which 0: A=attT -> x1_att (out tensor 0 ch1)
//                   which 1: A=attN -> x2_att (out tensor 1 ch1)
__global__ void k_attgemm(const float* __restrict__ attT, const float* __restrict__ attN,
                          const float* __restrict__ W, const float* __restrict__ bias,
                          float* __restrict__ out, float* __restrict__ stats) {
    const int z = blockIdx.z;
    const int b = z >> 1, which = z & 1;
    const float* A = which ? attN : attT;
    float* o = out + (size_t)which * (B_ * 2 * SD) + (size_t)b * (2 * SD) + SD;

    const int m0 = blockIdx.y * 16, n0 = blockIdx.x * 16;
    const int L = threadIdx.x, lo = L & 15, hi = L >> 4, koff = hi * 2;

    const v2f* arow = (const v2f*)(A + (size_t)(b * S_ + m0 + lo) * S_);
    const v2f* wrow = (const v2f*)(W + (size_t)(n0 + lo) * S_);

    v8f acc = {};
#pragma unroll 4
    for (int t = 0; t < 64; ++t) {
        v2f a = arow[(4 * t + koff) >> 1];
        v2f w = wrow[(4 * t + koff) >> 1];
        acc = __builtin_amdgcn_wmma_f32_16x16x4_f32(false, a, false, w,
                                                    (short)0, acc, false, false);
    }
    const float bb = bias[n0 + lo];
    float sum = 0.0f, ssq = 0.0f;
#pragma unroll
    for (int r = 0; r < 8; ++r) {
        float v = acc[r] + bb;
        sum += v; ssq += v * v;
        o[(size_t)(m0 + r + 8 * hi) * D_ + n0 + lo] = v;   // coalesced per half-wave
    }
#pragma unroll
    for (int off = 16; off; off >>= 1) {
        sum += __shfl_xor(sum, off, 32);
        ssq += __shfl_xor(ssq, off, 32);
    }
    if (L == 0) {
        const int ch = which ? 3 : 1;
        atomicAdd(&stats[2 * ch],     sum);
        atomicAdd(&stats[2 * ch + 1], ssq);
    }
}

// ---------------- BN stats for the raw channels (x1, x2) ----------------
__global__ void k_rawstats(const float* __restrict__ x1, const float* __restrict__ x2,
                           float* __restrict__ stats) {
    const int t = blockIdx.y;                 // 0: x1, 1: x2
    const float* src = t ? x2 : x1;
    float sum = 0.0f, ssq = 0.0f;
    const int base = blockIdx.x * 2048 + threadIdx.x;
#pragma unroll
    for (int k = 0; k < 8; ++k) { float v = src[base + 256 * k]; sum += v; ssq += v * v; }
#pragma unroll
    for (int off = 16; off; off >>= 1) {
        sum += __shfl_xor(sum, off, 32);
        ssq += __shfl_xor(ssq, off, 32);
    }
    if ((threadIdx.x & 31) == 0) {
        const int ch = t ? 2 : 0;
        atomicAdd(&stats[2 * ch],     sum);
        atomicAdd(&stats[2 * ch + 1], ssq);
    }
}

// ---------------- BN apply: fill ch0 from raw inputs, normalize ch1 in place ----------------
__global__ void k_bn(const float* __restrict__ x1, const float* __restrict__ x2,
                     const float* __restrict__ stats, const float* __restrict__ gamma,
                     const float* __restrict__ beta, float* __restrict__ out) {
    const float invN = 1.0f / (float)(B_ * SD);
#pragma unroll
    for (int k = 0; k < 4; ++k) {
        const int e   = blockIdx.x * 1024 + threadIdx.x + 256 * k;
        const int tp  = e >> 21;            // which output tensor (0:x1c, 1:x2c)
        const int r   = e & 2097151;
        const int b   = r >> 17;            // / (2*SD)
        const int c   = (r >> 16) & 1;      // channel
        const int pos = r & 65535;
        const int ch  = tp * 2 + c;
        const float mean = stats[2 * ch] * invN;
        const float var  = stats[2 * ch + 1] * invN - mean * mean;
        const float scale = rsqrtf(var + EPS_BN) * gamma[c];
        const float val = c ? out[e] : (tp ? x2[b * SD + pos] : x1[b * SD + pos]);
        out[e] = (val - mean) * scale + beta[c];
    }
}

extern "C" void kernel_launch(void* const* d_in, const int* in_sizes, int n_in,
                              void* d_out, int out_size, void* d_ws, size_t ws_size,
                              hipStream_t stream) {
    const float* x1    = (const float*)d_in[0];
    const float* x2    = (const float*)d_in[1];
    const float* W     = (const float*)d_in[2];
    const float* bias  = (const float*)d_in[3];
    const float* gamma = (const float*)d_in[4];
    const float* beta  = (const float*)d_in[5];
    float* out = (float*)d_out;
    float* ws  = (float*)d_ws;

    float* norms = ws;                 // N1 @0, N2 @4096
    float* attT  = ws + OFF_ATT_T;
    float* attN  = ws + OFF_ATT_N;
    float* stats = ws + OFF_STATS;

    hipLaunchKernelGGL(k_zero,     dim3(1),           dim3(32),  0, stream, stats, 8);
    hipLaunchKernelGGL(k_norms,    dim3(1024),        dim3(256), 0, stream, x1, x2, norms);
    hipLaunchKernelGGL(k_attn,     dim3(16, 16, B_),  dim3(32),  0, stream, x1, x2, norms, attT, attN);
    hipLaunchKernelGGL(k_rawstats, dim3(512, 2),      dim3(256), 0, stream, x1, x2, stats);
    hipLaunchKernelGGL(k_attgemm,  dim3(16, 16, B_*2),dim3(32),  0, stream, attT, attN, W, bias, out, stats);
    hipLaunchKernelGGL(k_bn,       dim3(4096),        dim3(256), 0, stream, x1, x2, stats, gamma, beta, out);
}